// THPAEncFR3_26723286516059
// MI455X (gfx1250) — compile-verified
//
#include <hip/hip_runtime.h>
#include <cmath>

typedef float v2f __attribute__((ext_vector_type(2)));
typedef float v8f __attribute__((ext_vector_type(8)));

#define DEVINL __device__ __forceinline__

static constexpr int  Wd = 64, Hd = 64, Dd = 64;
static constexpr int  WHD = Wd * Hd * Dd;          // 262144
// workspace offsets (in floats)
static constexpr size_t OFF_STATS1 = 0;            // 64  * 2  (mean,rstd) for xh instance norm
static constexpr size_t OFF_STATSZ = 128;          // 128 * 2  (mean,rstd) for z instance norm
static constexpr size_t OFF_KXY    = 384;          // 8*64*64
static constexpr size_t OFF_KZ     = 33152;        // 8*64
static constexpr size_t OFF_KY     = 33664;        // 8*64
static constexpr size_t OFF_G      = 34176;        // 128*32*64*64 (ghpa output g)

DEVINL float gelu_(float v) { return 0.5f * v * (1.0f + erff(v * 0.70710678118654752440f)); }

// 1D align_corners interp of a length-32 row sampled at integer output pos in [0,64)
DEVINL float lin1d(const float* __restrict__ p, int pos) {
  float f = (float)pos * (31.0f / 63.0f);
  int i0 = (int)f; if (i0 > 31) i0 = 31;
  int i1 = (i0 < 31) ? i0 + 1 : 31;
  float w = f - (float)i0;
  return p[i0] * (1.0f - w) + p[i1] * w;
}

// bilinear align_corners interp of a 32x32 grid at integer output pos (y,x) in [0,64)^2
DEVINL float bil2d(const float* __restrict__ p, int y, int x) {
  float fy = (float)y * (31.0f / 63.0f);
  float fx = (float)x * (31.0f / 63.0f);
  int y0 = (int)fy; if (y0 > 31) y0 = 31; int y1 = (y0 < 31) ? y0 + 1 : 31; float wy = fy - (float)y0;
  int x0 = (int)fx; if (x0 > 31) x0 = 31; int x1 = (x0 < 31) ? x0 + 1 : 31; float wx = fx - (float)x0;
  float a = p[y0 * 32 + x0] * (1.0f - wx) + p[y0 * 32 + x1] * wx;
  float b = p[y1 * 32 + x0] * (1.0f - wx) + p[y1 * 32 + x1] * wx;
  return a * (1.0f - wy) + b * wy;
}

// ---------------------------------------------------------------------------
// K1: instance-norm stats for xh channels (b in 0..1, c in 0..31)
// ---------------------------------------------------------------------------
__global__ __launch_bounds__(256) void k_stats_xh(const float* __restrict__ x, float* __restrict__ wsf) {
  int bc = blockIdx.x;                 // 0..63
  int b = bc >> 5, c = bc & 31;
  const float* base = x + (size_t)(b * 64 + c) * WHD;
  float s = 0.f, q = 0.f;
  for (int i = threadIdx.x; i < WHD; i += 256) { float v = base[i]; s += v; q += v * v; }
  __shared__ float rs[256], rq[256];
  rs[threadIdx.x] = s; rq[threadIdx.x] = q; __syncthreads();
  for (int st = 128; st > 0; st >>= 1) {
    if (threadIdx.x < st) { rs[threadIdx.x] += rs[threadIdx.x + st]; rq[threadIdx.x] += rq[threadIdx.x + st]; }
    __syncthreads();
  }
  if (threadIdx.x == 0) {
    float m = rs[0] * (1.0f / (float)WHD);
    float v = rq[0] * (1.0f / (float)WHD) - m * m;
    wsf[OFF_STATS1 + bc * 2]     = m;
    wsf[OFF_STATS1 + bc * 2 + 1] = rsqrtf(v + 1e-5f);
  }
}

// ---------------------------------------------------------------------------
// K2: build data-independent Hadamard kernels k_xy (8x64x64), kz (8x64), ky (8x64)
// ---------------------------------------------------------------------------
__global__ __launch_bounds__(256) void k_build_kernels(
    const float* __restrict__ p_xy, const float* __restrict__ p_zx, const float* __restrict__ p_zy,
    const float* __restrict__ xy_dw_w, const float* __restrict__ xy_dw_b,
    const float* __restrict__ xy_pw_w, const float* __restrict__ xy_pw_b,
    const float* __restrict__ zx_dw_w, const float* __restrict__ zx_dw_b,
    const float* __restrict__ zx_pw_w, const float* __restrict__ zx_pw_b,
    const float* __restrict__ zy_dw_w, const float* __restrict__ zy_dw_b,
    const float* __restrict__ zy_pw_w, const float* __restrict__ zy_pw_b,
    float* __restrict__ wsf) {
  int t = blockIdx.x * 256 + threadIdx.x;
  if (t < 32768) {  // k_xy[c][a][d]
    int c = t >> 12, rem = t & 4095, a = rem >> 6, d = rem & 63;
    float acc = xy_pw_b[c];
    for (int ci = 0; ci < 8; ++ci) {
      float dw = xy_dw_b[ci];
      for (int dy = -1; dy <= 1; ++dy)
        for (int dx = -1; dx <= 1; ++dx) {
          int ya = a + dy, xd = d + dx;
          if ((unsigned)ya < 64u && (unsigned)xd < 64u)
            dw += xy_dw_w[ci * 9 + (dy + 1) * 3 + (dx + 1)] * bil2d(p_xy + ci * 1024, ya, xd);
        }
      acc += xy_pw_w[c * 8 + ci] * gelu_(dw);
    }
    wsf[OFF_KXY + t] = acc;
  } else if (t < 33280) {  // kz[c][a]
    int j = t - 32768; int c = j >> 6, a = j & 63;
    float acc = zx_pw_b[c];
    for (int ci = 0; ci < 8; ++ci) {
      float dw = zx_dw_b[ci];
      for (int dt = -1; dt <= 1; ++dt) {
        int aa = a + dt;
        if ((unsigned)aa < 64u) dw += zx_dw_w[ci * 3 + dt + 1] * lin1d(p_zx + ci * 32, aa);
      }
      acc += zx_pw_w[c * 8 + ci] * gelu_(dw);
    }
    wsf[OFF_KZ + j] = acc;
  } else if (t < 33792) {  // ky[c][d]
    int j = t - 33280; int c = j >> 6, d = j & 63;
    float acc = zy_pw_b[c];
    for (int ci = 0; ci < 8; ++ci) {
      float dw = zy_dw_b[ci];
      for (int dt = -1; dt <= 1; ++dt) {
        int dd = d + dt;
        if ((unsigned)dd < 64u) dw += zy_dw_w[ci * 3 + dt + 1] * lin1d(p_zy + ci * 32, dd);
      }
      acc += zy_pw_w[c * 8 + ci] * gelu_(dw);
    }
    wsf[OFF_KY + j] = acc;
  }
}

// ---------------------------------------------------------------------------
// K3: fused GHPA. One 16x8 (w,d) tile per workgroup, halo 2. Writes g[n][c][w][d].
// ---------------------------------------------------------------------------
__global__ __launch_bounds__(256) void k_ghpa(
    const float* __restrict__ x,
    const float* __restrict__ n1_g, const float* __restrict__ n1_b,
    const float* __restrict__ n2_g, const float* __restrict__ n2_b,
    const float* __restrict__ dw_pw_w, const float* __restrict__ dw_pw_b,
    const float* __restrict__ dw_dw_w, const float* __restrict__ dw_dw_b,
    const float* __restrict__ ldw_dw_w, const float* __restrict__ ldw_dw_b,
    const float* __restrict__ ldw_pw_w, const float* __restrict__ ldw_pw_b,
    float* __restrict__ wsf) {
  constexpr int TW = 16, TD = 8, HWp = 20, HDp = 12, IWp = 18, IDp = 10;
  __shared__ float sm[15360];                 // 60 KB
  float* ln1 = sm;                            // 20*12*32 = 7680
  float* s4  = sm + 7680;                     // 20*12*8  = 1920
  float* zn  = sm + 9600;                     // 18*10*32 = 5760
  float* gu  = sm;                            // overlays ln1 (needs 16*8*32 = 4096)

  int tid = threadIdx.x;
  int bid = blockIdx.x;                       // 128 * 32 tiles
  int n = bid >> 5, tt = bid & 31;
  int w0 = (tt >> 3) * TW, d0 = (tt & 7) * TD;
  int h = n >> 1, b = n & 1;
  const float* st1 = wsf + OFF_STATS1;
  const float* kxy = wsf + OFF_KXY;
  const float* kz  = wsf + OFF_KZ;
  const float* ky  = wsf + OFF_KY;
  float* g = wsf + OFF_G;

  // Phase A: load instance-normed xh into LDS (zeros outside image)
  for (int i = tid; i < HWp * HDp * 32; i += 256) {
    int c = i / (HWp * HDp), p = i - c * (HWp * HDp);
    int pw = p / HDp, pd = p - pw * HDp;
    int w = w0 + pw - 2, d = d0 + pd - 2;
    float v = 0.f;
    if ((unsigned)w < 64u && (unsigned)d < 64u) {
      float raw = x[(size_t)((b * 64 + c) * 64 + w) * 4096 + h * 64 + d];
      v = (raw - st1[(b * 32 + c) * 2]) * st1[(b * 32 + c) * 2 + 1];
    }
    ln1[p * 32 + c] = v;
  }
  __syncthreads();

  // Phase B: per-pixel LayerNorm over 32 channels (only for in-image pixels)
  if (tid < HWp * HDp) {
    int p = tid, pw = p / HDp, pd = p - pw * HDp;
    int w = w0 + pw - 2, d = d0 + pd - 2;
    if ((unsigned)w < 64u && (unsigned)d < 64u) {
      float mu = 0.f;
      #pragma unroll
      for (int c = 0; c < 32; ++c) mu += ln1[p * 32 + c];
      mu *= (1.0f / 32.0f);
      float var = 0.f;
      #pragma unroll
      for (int c = 0; c < 32; ++c) { float t2 = ln1[p * 32 + c] - mu; var += t2 * t2; }
      float rs = rsqrtf(var * (1.0f / 32.0f) + 1e-6f);
      #pragma unroll
      for (int c = 0; c < 32; ++c)
        ln1[p * 32 + c] = (ln1[p * 32 + c] - mu) * rs * n1_g[c] + n1_b[c];
    }
  }
  __syncthreads();

  // Phase C: s4 = gelu(pointwise 8->8 of channels 24..31)
  for (int i = tid; i < HWp * HDp * 8; i += 256) {
    int c4 = i & 7, p = i >> 3;
    int pw = p / HDp, pd = p - pw * HDp;
    int w = w0 + pw - 2, d = d0 + pd - 2;
    float v = 0.f;
    if ((unsigned)w < 64u && (unsigned)d < 64u) {
      float a = dw_pw_b[c4];
      #pragma unroll
      for (int ci = 0; ci < 8; ++ci) a += dw_pw_w[c4 * 8 + ci] * ln1[p * 32 + 24 + ci];
      v = gelu_(a);
    }
    s4[p * 8 + c4] = v;
  }
  __syncthreads();

  // Phase D: branch Hadamards + dw conv of s4, then LayerNorm2 -> zn (inner 18x10)
  if (tid < IWp * IDp) {
    int q = tid, qw = q / IDp, qd = q - qw * IDp;
    int pw = qw + 1, pd = qd + 1, p = pw * HDp + pd;
    int w = w0 + qw - 1, d = d0 + qd - 1;
    if ((unsigned)w < 64u && (unsigned)d < 64u) {
      float zl[32];
      #pragma unroll
      for (int c = 0; c < 8; ++c) zl[c]      = ln1[p * 32 + c]      * kxy[(c * 64 + w) * 64 + d];
      #pragma unroll
      for (int c = 0; c < 8; ++c) zl[8 + c]  = ln1[p * 32 + 8 + c]  * kz[c * 64 + w];
      #pragma unroll
      for (int c = 0; c < 8; ++c) zl[16 + c] = ln1[p * 32 + 16 + c] * ky[c * 64 + d];
      #pragma unroll
      for (int c4 = 0; c4 < 8; ++c4) {
        float a = dw_dw_b[c4];
        #pragma unroll
        for (int dy = -1; dy <= 1; ++dy)
          #pragma unroll
          for (int dx = -1; dx <= 1; ++dx)
            a += dw_dw_w[c4 * 9 + (dy + 1) * 3 + (dx + 1)] * s4[((pw + dy) * HDp + (pd + dx)) * 8 + c4];
        zl[24 + c4] = a;
      }
      float mu = 0.f;
      #pragma unroll
      for (int c = 0; c < 32; ++c) mu += zl[c];
      mu *= (1.0f / 32.0f);
      float var = 0.f;
      #pragma unroll
      for (int c = 0; c < 32; ++c) { float t2 = zl[c] - mu; var += t2 * t2; }
      float rs = rsqrtf(var * (1.0f / 32.0f) + 1e-6f);
      #pragma unroll
      for (int c = 0; c < 32; ++c)
        zn[q * 32 + c] = (zl[c] - mu) * rs * n2_g[c] + n2_b[c];
    } else {
      #pragma unroll
      for (int c = 0; c < 32; ++c) zn[q * 32 + c] = 0.f;
    }
  }
  __syncthreads();

  // Phase E1: gu = gelu(depthwise 3x3 of zn) on the 16x8 core
  for (int i = tid; i < TW * TD * 32; i += 256) {
    int c = i / (TW * TD), p = i - c * (TW * TD);
    int ow = p / TD, od = p - ow * TD;
    int qw = ow + 1, qd = od + 1;
    float u = ldw_dw_b[c];
    #pragma unroll
    for (int dy = -1; dy <= 1; ++dy)
      #pragma unroll
      for (int dx = -1; dx <= 1; ++dx)
        u += ldw_dw_w[c * 9 + (dy + 1) * 3 + (dx + 1)] * zn[((qw + dy) * IDp + (qd + dx)) * 32 + c];
    gu[p * 32 + c] = gelu_(u);
  }
  __syncthreads();

  // Phase E2: 1x1 conv 32->32 and store g
  for (int i = tid; i < TW * TD * 32; i += 256) {
    int c = i / (TW * TD), p = i - c * (TW * TD);
    int ow = p / TD, od = p - ow * TD;
    float acc = ldw_pw_b[c];
    #pragma unroll
    for (int ci = 0; ci < 32; ++ci) acc += ldw_pw_w[c * 32 + ci] * gu[p * 32 + ci];
    g[(size_t)((n * 32 + c) * 64 + (w0 + ow)) * 64 + (d0 + od)] = acc;
  }
}

// ---------------------------------------------------------------------------
// K4: instance-norm stats for z = concat(whd = g + flip_h(g), xres)
// ---------------------------------------------------------------------------
__global__ __launch_bounds__(256) void k_stats_z(const float* __restrict__ x, float* __restrict__ wsf) {
  int bc = blockIdx.x;                // 0..127
  int b = bc >> 6, c = bc & 63;
  const float* g = wsf + OFF_G;
  float s = 0.f, q = 0.f;
  for (int i = threadIdx.x; i < WHD; i += 256) {
    float v;
    if (c < 32) {
      int w = i >> 12, h = (i >> 6) & 63, d = i & 63;
      int n1 = h * 2 + b, n2 = (63 - h) * 2 + b;
      v = g[(size_t)((n1 * 32 + c) * 64 + w) * 64 + d] +
          g[(size_t)((n2 * 32 + c) * 64 + w) * 64 + d];
    } else {
      v = x[(size_t)(b * 64 + c) * WHD + i];
    }
    s += v; q += v * v;
  }
  __shared__ float rs[256], rq[256];
  rs[threadIdx.x] = s; rq[threadIdx.x] = q; __syncthreads();
  for (int st = 128; st > 0; st >>= 1) {
    if (threadIdx.x < st) { rs[threadIdx.x] += rs[threadIdx.x + st]; rq[threadIdx.x] += rq[threadIdx.x + st]; }
    __syncthreads();
  }
  if (threadIdx.x == 0) {
    float m = rs[0] * (1.0f / (float)WHD);
    float v = rq[0] * (1.0f / (float)WHD) - m * m;
    wsf[OFF_STATSZ + bc * 2]     = m;
    wsf[OFF_STATSZ + bc * 2 + 1] = rsqrtf(v + 1e-5f);
  }
}

// ---------------------------------------------------------------------------
// K5: per-voxel MLP 64 -> 128 -> gelu -> 64 using V_WMMA_F32_16X16X4_F32.
// 4 waves/WG; each wave owns 16 consecutive voxels (one d-run of a (b,w,h) row).
// ---------------------------------------------------------------------------
__global__ __launch_bounds__(128) void k_mlp_wmma(
    const float* __restrict__ x,
    const float* __restrict__ fc1_w, const float* __restrict__ fc1_b,
    const float* __restrict__ fc2_w, const float* __restrict__ fc2_b,
    const float* __restrict__ wsf, float* __restrict__ out) {
  __shared__ float sm[12800];                  // 51.2 KB
  int tid = threadIdx.x, lane = tid & 31, wv = tid >> 5;
  int bid = blockIdx.x;                        // 8192 = 2*64*64 (b,w,h) rows
  int h = bid & 63, w = (bid >> 6) & 63, b = bid >> 12;
  const float* g   = wsf + OFF_G;
  const float* stz = wsf + OFF_STATSZ;
  float* Hs   = sm + wv * 2112;                // 16 x 132 (padded)
  float* zt   = sm + 8448 + wv * 1088;         // 16 x 68 (padded)
  float* outs = sm + 8448;                     // 64 x 68, overlays the zt region
  int d0 = wv * 16;

  // Stage normalized z tile (16 voxels x 64 channels) into LDS
  for (int it = 0; it < 32; ++it) {
    int e = it * 32 + lane;
    int c = e >> 4, vox = e & 15, d = d0 + vox;
    float v;
    if (c < 32) {
      int n1 = h * 2 + b, n2 = (63 - h) * 2 + b;
      v = g[(size_t)((n1 * 32 + c) * 64 + w) * 64 + d] +
          g[(size_t)((n2 * 32 + c) * 64 + w) * 64 + d];
    } else {
      v = x[(size_t)((b * 64 + c) * 64 + w) * 4096 + h * 64 + d];
    }
    int sc = (b * 64 + c) * 2;
    zt[vox * 68 + c] = (v - stz[sc]) * stz[sc + 1];
  }
  __syncthreads();

  int row   = lane & 15;               // A-matrix M row / B-matrix N col owned by this lane
  int khalf = (lane >> 4) << 1;        // lanes 16..31 hold K = {2,3} of each k-step

  // fc1: D(16x16) += A(16x4: voxels x chans) * B(4x16: chans x hidden), K = 64
  for (int j = 0; j < 8; ++j) {
    v8f acc = {0.f, 0.f, 0.f, 0.f, 0.f, 0.f, 0.f, 0.f};
    int o = j * 16 + row;
    #pragma unroll
    for (int kk = 0; kk < 16; ++kk) {
      int k0 = kk * 4 + khalf;
      v2f a;  a[0]  = zt[row * 68 + k0];      a[1]  = zt[row * 68 + k0 + 1];
      v2f bm; bm[0] = fc1_w[o * 64 + k0];     bm[1] = fc1_w[o * 64 + k0 + 1];
      acc = __builtin_amdgcn_wmma_f32_16x16x4_f32(false, a, false, bm, (short)0, acc,
                                                  false, false);
    }
    float bias = fc1_b[o];
    #pragma unroll
    for (int r = 0; r < 8; ++r) {
      int mrow = (lane < 16) ? r : r + 8;    // C layout: vgpr r -> M=r (lo) / M=r+8 (hi)
      Hs[mrow * 132 + j * 16 + row] = gelu_(acc[r] + bias);
    }
  }
  __syncthreads();

  // fc2: K = 128, N = 64
  for (int jo = 0; jo < 4; ++jo) {
    v8f acc = {0.f, 0.f, 0.f, 0.f, 0.f, 0.f, 0.f, 0.f};
    int o = jo * 16 + row;
    #pragma unroll
    for (int kk = 0; kk < 32; ++kk) {
      int k0 = kk * 4 + khalf;
      v2f a;  a[0]  = Hs[row * 132 + k0];     a[1]  = Hs[row * 132 + k0 + 1];
      v2f bm; bm[0] = fc2_w[o * 128 + k0];    bm[1] = fc2_w[o * 128 + k0 + 1];
      acc = __builtin_amdgcn_wmma_f32_16x16x4_f32(false, a, false, bm, (short)0, acc,
                                                  false, false);
    }
    float bias = fc2_b[o];
    #pragma unroll
    for (int r = 0; r < 8; ++r) {
      int mrow = (lane < 16) ? r : r + 8;
      outs[(wv * 16 + mrow) * 68 + jo * 16 + row] = acc[r] + bias;  // [vox][oc]
    }
  }
  __syncthreads();

  // Coalesced store: 64 out-channels x 64 contiguous d for this (b,w,h) row
  size_t obase = (size_t)(b * 64) * WHD + (size_t)w * 4096 + (size_t)h * 64;
  for (int it = 0; it < 32; ++it) {
    int idx = it * 128 + tid;
    int oc = idx >> 6, vx = idx & 63;
    out[obase + (size_t)oc * WHD + vx] = outs[vx * 68 + oc];
  }
}

// ---------------------------------------------------------------------------
extern "C" void kernel_launch(void* const* d_in, const int* in_sizes, int n_in,
                              void* d_out, int out_size, void* d_ws, size_t ws_size,
                              hipStream_t stream) {
  (void)in_sizes; (void)n_in; (void)out_size; (void)ws_size;
  const float* x        = (const float*)d_in[0];
  const float* n1_g     = (const float*)d_in[1];
  const float* n1_b     = (const float*)d_in[2];
  const float* n2_g     = (const float*)d_in[3];
  const float* n2_b     = (const float*)d_in[4];
  const float* p_xy     = (const float*)d_in[5];
  const float* p_zx     = (const float*)d_in[6];
  const float* p_zy     = (const float*)d_in[7];
  const float* xy_dw_w  = (const float*)d_in[8];
  const float* xy_dw_b  = (const float*)d_in[9];
  const float* xy_pw_w  = (const float*)d_in[10];
  const float* xy_pw_b  = (const float*)d_in[11];
  const float* zx_dw_w  = (const float*)d_in[12];
  const float* zx_dw_b  = (const float*)d_in[13];
  const float* zx_pw_w  = (const float*)d_in[14];
  const float* zx_pw_b  = (const float*)d_in[15];
  const float* zy_dw_w  = (const float*)d_in[16];
  const float* zy_dw_b  = (const float*)d_in[17];
  const float* zy_pw_w  = (const float*)d_in[18];
  const float* zy_pw_b  = (const float*)d_in[19];
  const float* dw_pw_w  = (const float*)d_in[20];
  const float* dw_pw_b  = (const float*)d_in[21];
  const float* dw_dw_w  = (const float*)d_in[22];
  const float* dw_dw_b  = (const float*)d_in[23];
  const float* ldw_dw_w = (const float*)d_in[24];
  const float* ldw_dw_b = (const float*)d_in[25];
  const float* ldw_pw_w = (const float*)d_in[26];
  const float* ldw_pw_b = (const float*)d_in[27];
  const float* fc1_w    = (const float*)d_in[28];
  const float* fc1_b    = (const float*)d_in[29];
  const float* fc2_w    = (const float*)d_in[30];
  const float* fc2_b    = (const float*)d_in[31];
  float* wsf = (float*)d_ws;
  float* out = (float*)d_out;

  k_stats_xh<<<64, 256, 0, stream>>>(x, wsf);
  k_build_kernels<<<132, 256, 0, stream>>>(p_xy, p_zx, p_zy,
      xy_dw_w, xy_dw_b, xy_pw_w, xy_pw_b,
      zx_dw_w, zx_dw_b, zx_pw_w, zx_pw_b,
      zy_dw_w, zy_dw_b, zy_pw_w, zy_pw_b, wsf);
  k_ghpa<<<4096, 256, 0, stream>>>(x, n1_g, n1_b, n2_g, n2_b,
      dw_pw_w, dw_pw_b, dw_dw_w, dw_dw_b,
      ldw_dw_w, ldw_dw_b, ldw_pw_w, ldw_pw_b, wsf);
  k_stats_z<<<128, 256, 0, stream>>>(x, wsf);
  k_mlp_wmma<<<8192, 128, 0, stream>>>(x, fc1_w, fc1_b, fc2_w, fc2_b, wsf, out);
}